// Cvx_ShortestPathNet_55740085567735
// MI455X (gfx1250) — compile-verified
//
#include <hip/hip_runtime.h>

typedef __attribute__((ext_vector_type(2))) float v2f;
typedef __attribute__((ext_vector_type(8))) float v8f;

#define BATCH 256
#define CTX   64
#define HID   640
#define N1    400
#define N2    760
#define LDW   768          // padded leading dim for 760-wide buffers
#define NS_ITERS  30       // Newton-Schulz iterations for pinv(L)
#define DYK_ITERS 100
#define LEAKY 0.1f

// ---------------------------------------------------------------------------
// One wave computes MT 16x16 tiles stacked in M (MT=1 or 2) of
//   C = abScale*(A@B[^T]) + eScale*E + biasScale*bias, optional LeakyReLU.
// fp32 WMMA 16x16x4. All strides and K are compile-time template parameters
// so every load is base-pointer + immediate offset (24-bit IOFFSET covers the
// largest ka*LDB*4 ~ 1.9MB) with one pointer increment per loop iteration.
//
// MT=2 shares each B fragment across two WMMAs (two M-tiles), doubling wmma
// per B load; used for all M=256 batch GEMMs (M%32==0).
//
// GUARD (ragged N=760 outputs only): unconditional loads from a clamped
// address + v_cndmask zeroing -- no exec manipulation, EXEC stays all-ones
// at every WMMA as the ISA requires.
//
// Fragment layouts per CDNA5 ISA (wave32):
//   A 16x4 : lanes 0-15 hold M=lane, v0=K0 v1=K1; lanes 16-31 M=lane-16, K2/K3
//   B 4x16 : lanes 0-15 hold N=lane, v0=K0 v1=K1; lanes 16-31 N=lane-16, K2/K3
//   C 16x16: VGPR r, lanes 0-15 -> M=r, lanes 16-31 -> M=r+8; N=lane&15
// ---------------------------------------------------------------------------
template <bool TRANSB, bool GUARD, int MT, int LDA, int LDB, int LDC, int LDE,
          int KDIM>
__global__ __launch_bounds__(32) void gemm16(
    const float* __restrict__ Am,
    const float* __restrict__ Bm,
    float* __restrict__ Cm,
    int Ndim,
    const float* __restrict__ Em, float eScale,
    float abScale,
    const float* __restrict__ bias, float biasScale,
    int leaky)
{
    const int lane = threadIdx.x;
    const int hi   = lane >> 4;      // 0: K0/K1 half, 1: K2/K3 half
    const int lr   = lane & 15;
    const int m0   = blockIdx.x * (16 * MT);
    const int n0   = blockIdx.y << 4;
    const int n    = n0 + lr;
    const bool nin = GUARD ? (n < Ndim) : true;
    const int  nc  = (GUARD && !nin) ? 0 : n;   // clamped, always in-bounds

    const float* arow0 = Am + (size_t)(m0 + lr) * LDA;
    const float* arow1 = arow0 + (size_t)16 * LDA;           // MT==2 only
    const float* bptr  = TRANSB ? (Bm + (size_t)nc * LDB) : (Bm + nc);

    v8f acc0 = {};
    v8f acc1 = {};
    for (int k = 0; k < KDIM; k += 8) {
        const int ka = k + (hi << 1);
        v2f a0 = *(const v2f*)(arow0 + ka);          // global_load_b64
        v2f a1 = *(const v2f*)(arow0 + ka + 4);
        v2f a2, a3;
        if (MT == 2) {
            a2 = *(const v2f*)(arow1 + ka);
            a3 = *(const v2f*)(arow1 + ka + 4);
        }
        v2f b0, b1;
        if (TRANSB) {
            b0 = *(const v2f*)(bptr + ka);           // global_load_b64
            b1 = *(const v2f*)(bptr + ka + 4);
        } else {                                     // imm-offset b32 loads
            b0.x = bptr[(size_t)ka * LDB];
            b0.y = bptr[(size_t)(ka + 1) * LDB];
            b1.x = bptr[(size_t)(ka + 4) * LDB];
            b1.y = bptr[(size_t)(ka + 5) * LDB];
        }
        if (GUARD) {                                 // v_cndmask, no exec ops
            b0.x = nin ? b0.x : 0.f;  b0.y = nin ? b0.y : 0.f;
            b1.x = nin ? b1.x : 0.f;  b1.y = nin ? b1.y : 0.f;
        }
        __builtin_prefetch(arow0 + ka + 64, 0, 3);   // global_prefetch_b8
        if (MT == 2) __builtin_prefetch(arow1 + ka + 64, 0, 3);
        // (neg_a, A, neg_b, B, c_mod, C, reuse_a, reuse_b)
        acc0 = __builtin_amdgcn_wmma_f32_16x16x4_f32(
            false, a0, false, b0, (short)0, acc0, false, false);
        acc0 = __builtin_amdgcn_wmma_f32_16x16x4_f32(
            false, a1, false, b1, (short)0, acc0, false, false);
        if (MT == 2) {
            acc1 = __builtin_amdgcn_wmma_f32_16x16x4_f32(
                false, a2, false, b0, (short)0, acc1, false, false);
            acc1 = __builtin_amdgcn_wmma_f32_16x16x4_f32(
                false, a3, false, b1, (short)0, acc1, false, false);
        }
    }

    if (nin) {
        const float bv = bias ? biasScale * bias[n] : 0.f;
        auto epi = [&](const v8f& a, int mbase) {
#pragma unroll
            for (int r = 0; r < 8; ++r) {
                const int mr = mbase + r + (hi << 3);
                float v = abScale * a[r] + bv;
                if (Em)   v += eScale * Em[(size_t)mr * LDE + n];
                if (leaky) v = (v > 0.f) ? v : LEAKY * v;
                Cm[(size_t)mr * LDC + n] = v;
            }
        };
        epi(acc0, m0);
        if (MT == 2) epi(acc1, m0 + 16);
    }
}

// --------------------------- elementwise helpers ---------------------------
__global__ void k_zero(float* p, int n) {
    int i = blockIdx.x * blockDim.x + threadIdx.x;
    if (i < n) p[i] = 0.f;
}
__global__ void k_scale(float* d, const float* s, float a, int n) {
    int i = blockIdx.x * blockDim.x + threadIdx.x;
    if (i < n) d[i] = a * s[i];
}
// Dykstra init: y = t = -w  (p = q = 0 from the arena zero-fill)
__global__ void k_init(float* y, float* t, const float* w, int n) {
    int i = blockIdx.x * blockDim.x + threadIdx.x;
    if (i < n) { float v = -w[i]; y[i] = v; t[i] = v; }
}
// Fused Dykstra tail + next-iteration head:
//   p = t - yp; t2 = yp + q; y = max(t2,0); q = t2 - y; t = y + p
__global__ void k_postpre(float* p, float* q, float* y, float* t,
                          const float* yp, int n) {
    int i = blockIdx.x * blockDim.x + threadIdx.x;
    if (i < n) {
        float tv = t[i], ypv = yp[i], qv = q[i];
        float pn = tv - ypv;
        float t2 = ypv + qv;
        float yn = fmaxf(t2, 0.f);
        p[i] = pn;
        y[i] = yn;
        q[i] = t2 - yn;
        t[i] = yn + pn;
    }
}
__global__ void k_pack(float* out, const float* y, int rows, int cols, int ld) {
    int i = blockIdx.x * blockDim.x + threadIdx.x;
    if (i < rows * cols) {
        int r = i / cols, c = i - r * cols;
        out[i] = y[(size_t)r * ld + c];
    }
}

// ---------------------------------------------------------------------------
extern "C" void kernel_launch(void* const* d_in, const int* in_sizes, int n_in,
                              void* d_out, int out_size, void* d_ws, size_t ws_size,
                              hipStream_t stream)
{
    (void)in_sizes; (void)n_in; (void)out_size; (void)ws_size;
    const float* din = (const float*)d_in[0];   // [256,64]
    const float* W1  = (const float*)d_in[1];   // [64,640]
    const float* b1  = (const float*)d_in[2];   // [640]
    const float* W2  = (const float*)d_in[3];   // [640,760]
    const float* b2  = (const float*)d_in[4];   // [760]
    const float* Ain = (const float*)d_in[5];   // [400,760]
    const float* beq = (const float*)d_in[6];   // [400]

    float* ws = (float*)d_ws;
    float* h     = ws; ws += BATCH * HID;   // 256x640
    float* L     = ws; ws += N1 * N1;       // 400x400  A@A^T
    float* X     = ws; ws += N1 * N1;       // Newton-Schulz iterate -> pinv(L)
    float* X2    = ws; ws += N1 * N1;
    float* T1    = ws; ws += N1 * N1;
    float* resid = ws; ws += BATCH * N1;    // 256x400
    float* z     = ws; ws += BATCH * N1;    // 256x400
    // padded (ld=768) buffers, contiguous so one zero-fill covers them all:
    float* w  = ws; ws += BATCH * LDW;
    float* t  = ws; ws += BATCH * LDW;
    float* yp = ws; ws += BATCH * LDW;
    float* y  = ws; ws += BATCH * LDW;
    float* p  = ws; ws += BATCH * LDW;
    float* q  = ws; ws += BATCH * LDW;

    const dim3 blk(32);
    const int EW = 256;
    const int padN = 6 * BATCH * LDW;
    k_zero<<<(padN + EW - 1) / EW, EW, 0, stream>>>(w, padN);

    // h = LeakyReLU(d @ W1 + b1)            (N=640 exact; MT=2)
    gemm16<false, false, 2, CTX, HID, HID, 1, CTX>
        <<<dim3(BATCH / 32, HID / 16), blk, 0, stream>>>(
        din, W1, h, HID, nullptr, 0.f, 1.f, b1, 1.f, 1);
    // w = h @ W2 + b2                       (ragged N=760; MT=2)
    gemm16<false, true, 2, HID, N2, LDW, 1, HID>
        <<<dim3(BATCH / 32, (N2 + 15) / 16), blk, 0, stream>>>(
        h, W2, w, N2, nullptr, 0.f, 1.f, b2, 1.f, 0);
    // L = A @ A^T (grid Laplacian)          (N=400 exact; 25 M-tiles -> MT=1)
    gemm16<true, false, 1, N2, N2, N1, 1, N2>
        <<<dim3(N1 / 16, N1 / 16), blk, 0, stream>>>(
        Ain, Ain, L, N1, nullptr, 0.f, 1.f, nullptr, 0.f, 0);

    // Newton-Schulz for pinv(L): X0 = 0.02*L (0.02 < 2/lambda_max^2, lmax<=8)
    k_scale<<<(N1 * N1 + EW - 1) / EW, EW, 0, stream>>>(X, L, 0.02f, N1 * N1);
    for (int it = 0; it < NS_ITERS; ++it) {
        // T1 = L @ X
        gemm16<false, false, 1, N1, N1, N1, 1, N1>
            <<<dim3(N1 / 16, N1 / 16), blk, 0, stream>>>(
            L, X, T1, N1, nullptr, 0.f, 1.f, nullptr, 0.f, 0);
        // X2 = 2*X - X @ T1   (epilogue-fused)
        gemm16<false, false, 1, N1, N1, N1, N1, N1>
            <<<dim3(N1 / 16, N1 / 16), blk, 0, stream>>>(
            X, T1, X2, N1, X, 2.f, -1.f, nullptr, 0.f, 0);
        float* tmp = X; X = X2; X2 = tmp;
    }
    // X now holds M = pinv(A A^T)

    // Dykstra init: y = t = -w, p = q = 0
    k_init<<<(BATCH * LDW + EW - 1) / EW, EW, 0, stream>>>(y, t, w, BATCH * LDW);

    for (int it = 0; it < DYK_ITERS; ++it) {
        // resid = t @ A^T - b_eq            (N=400 exact; MT=2)
        gemm16<true, false, 2, LDW, N2, N1, 1, N2>
            <<<dim3(BATCH / 32, N1 / 16), blk, 0, stream>>>(
            t, Ain, resid, N1, nullptr, 0.f, 1.f, beq, -1.f, 0);
        // z = resid @ M                     (N=400 exact; MT=2)
        gemm16<false, false, 2, N1, N1, N1, 1, N1>
            <<<dim3(BATCH / 32, N1 / 16), blk, 0, stream>>>(
            resid, X, z, N1, nullptr, 0.f, 1.f, nullptr, 0.f, 0);
        // yp = t - z @ A   (epilogue-fused) (ragged N=760; MT=2)
        gemm16<false, true, 2, N1, N2, LDW, LDW, N1>
            <<<dim3(BATCH / 32, (N2 + 15) / 16), blk, 0, stream>>>(
            z, Ain, yp, N2, t, 1.f, -1.f, nullptr, 0.f, 0);
        // p = t - yp; t2 = yp + q; y = max(t2,0); q = t2 - y; t = y + p
        k_postpre<<<(BATCH * LDW + EW - 1) / EW, EW, 0, stream>>>(
            p, q, y, t, yp, BATCH * LDW);
    }

    k_pack<<<(BATCH * N2 + EW - 1) / EW, EW, 0, stream>>>(
        (float*)d_out, y, BATCH, N2, LDW);
}